// EDDFNModel_26482768347279
// MI455X (gfx1250) — compile-verified
//
#include <hip/hip_runtime.h>
#include <hip/hip_bf16.h>
#include <math.h>

#define BDIM 256
#define LDIM 512
#define EDIM 768
#define HDIM 384
#define NDOM 9
#define SLOPE 0.01f
#define LN_EPS 1e-5f

typedef __attribute__((ext_vector_type(2))) float v2f;
typedef __attribute__((ext_vector_type(8))) float v8f;

__device__ __forceinline__ float leaky(float x) { return x > 0.f ? x : SLOPE * x; }

// ---------------------------------------------------------------------------
// Kernel 1: attention pooling.  One workgroup per batch sample b.
// Pass 1: scores[l] = dot(input[b,l,:], w_att) -- b128 loads, wave per row.
// Softmax over L in LDS.
// Pass 2: bert[b,:] = sum_l p[l]*input[b,l,:] -- 192 threads own one float4
// column group each (fully-coalesced global_load_b128, pass-2 data L2-hot).
// ---------------------------------------------------------------------------
__global__ void attn_pool(const float* __restrict__ input,
                          const float* __restrict__ w_att,
                          float* __restrict__ bert) {
    __shared__ float4 wsh4[EDIM / 4];     // 192 float4
    __shared__ float  sc[LDIM];
    __shared__ float  red[256];
    const int tid  = threadIdx.x;
    const int b    = blockIdx.x;
    const int wave = tid >> 5;
    const int lane = tid & 31;

    const float4* wv4 = (const float4*)w_att;
    for (int i = tid; i < EDIM / 4; i += 256) wsh4[i] = wv4[i];
    __syncthreads();

    const float*  base  = input + (size_t)b * LDIM * EDIM;
    const float4* base4 = (const float4*)base;

    for (int l = wave; l < LDIM; l += 8) {
        const float4* row4 = base4 + (size_t)l * (EDIM / 4);
        float p = 0.f;
#pragma unroll
        for (int i = 0; i < EDIM / 128; ++i) {        // 6 iterations
            const int idx = lane + 32 * i;
            const float4 x = row4[idx];
            const float4 w = wsh4[idx];
            p += x.x * w.x + x.y * w.y + x.z * w.z + x.w * w.w;
        }
        for (int m = 16; m >= 1; m >>= 1) p += __shfl_xor(p, m, 32);
        if (lane == 0) sc[l] = p;
    }
    __syncthreads();

    // softmax over L
    float v = -3.4e38f;
    for (int l = tid; l < LDIM; l += 256) v = fmaxf(v, sc[l]);
    red[tid] = v; __syncthreads();
    for (int off = 128; off > 0; off >>= 1) {
        if (tid < off) red[tid] = fmaxf(red[tid], red[tid + off]);
        __syncthreads();
    }
    const float mx = red[0]; __syncthreads();
    float ps = 0.f;
    for (int l = tid; l < LDIM; l += 256) { float e0 = __expf(sc[l] - mx); sc[l] = e0; ps += e0; }
    red[tid] = ps; __syncthreads();
    for (int off = 128; off > 0; off >>= 1) {
        if (tid < off) red[tid] += red[tid + off];
        __syncthreads();
    }
    const float inv = 1.f / red[0]; __syncthreads();
    for (int l = tid; l < LDIM; l += 256) sc[l] *= inv;
    __syncthreads();

    // weighted sum: threads 0..191 each own one float4 column group
    if (tid < EDIM / 4) {
        float a0 = 0.f, a1 = 0.f, a2 = 0.f, a3 = 0.f;
#pragma unroll 4
        for (int l = 0; l < LDIM; ++l) {
            const float  p = sc[l];
            const float4 x = base4[(size_t)l * (EDIM / 4) + tid];
            a0 += p * x.x; a1 += p * x.y; a2 += p * x.z; a3 += p * x.w;
        }
        float4 r; r.x = a0; r.y = a1; r.z = a2; r.w = a3;
        ((float4*)(bert + (size_t)b * EDIM))[tid] = r;
    }
}

// ---------------------------------------------------------------------------
// Kernel 2: generic fused  C = LayerNorm(LeakyReLU(A[16-row stripe] @ W + b))
// using V_WMMA_F32_16X16X4_F32.  grid.x = M/16, block = 256 (8 waves).
// K, N are compile-time so all A/B loads use immediate offsets from a single
// base pointer -> independent address chains -> deep load/WMMA pipelining.
// ---------------------------------------------------------------------------
template <int K, int N>
__global__ void gemm_bias_leaky_ln(const float* __restrict__ Amat, int lda,
                                   const float* __restrict__ Wm,
                                   const float* __restrict__ bias,
                                   const float* __restrict__ gamma,
                                   const float* __restrict__ beta,
                                   float* __restrict__ Cm, int ldc) {
    __shared__ float hbuf[16 * N];       // 16 x N stripe
    const int tid  = threadIdx.x;
    const int wave = tid >> 5;
    const int lane = tid & 31;
    const int m0   = blockIdx.x * 16;
    const int lrow = lane & 15;
    const int kb   = (lane >> 4) << 1;   // 0 for lanes 0-15, 2 for lanes 16-31

    const float* Arow = Amat + (size_t)(m0 + lrow) * lda + kb;

    for (int t = wave; t < N / 16; t += 8) {
        const int col = (t << 4) + lrow;
        const float* Bp = Wm + (size_t)kb * N + col;
        v8f acc = {0.f, 0.f, 0.f, 0.f, 0.f, 0.f, 0.f, 0.f};
#pragma unroll 16
        for (int k0 = 0; k0 < K; k0 += 4) {
            // A 16x4: lane -> (M=lane&15, K = k0+kb, k0+kb+1)  (contiguous pair)
            v2f a = *(const v2f*)(Arow + k0);
            // B 4x16: lane -> (K = k0+kb / +1, N = col)  (immediate offsets)
            v2f bb;
            bb.x = Bp[(size_t)k0 * N];
            bb.y = Bp[(size_t)(k0 + 1) * N];
            acc = __builtin_amdgcn_wmma_f32_16x16x4_f32(
                false, a, false, bb, (short)0, acc, false, false);
        }
        // C/D: VGPR r -> row r + (lane>=16 ? 8 : 0), col = lane&15
        const float bcol  = bias[col];
        const int   rbase = (lane >> 4) << 3;
#pragma unroll
        for (int r = 0; r < 8; ++r) {
            hbuf[(rbase + r) * N + col] = leaky(acc[r] + bcol);
        }
    }
    __syncthreads();

    // LayerNorm per row: 16 threads per row (tid>>4 = row, tid&15 = sub-lane)
    const int row = tid >> 4;
    const int idx = tid & 15;
    float s = 0.f, sq = 0.f;
#pragma unroll
    for (int j = 0; j < N / 16; ++j) {
        const float x = hbuf[row * N + idx + 16 * j];
        s += x; sq += x * x;
    }
    for (int m = 8; m >= 1; m >>= 1) { s += __shfl_xor(s, m, 32); sq += __shfl_xor(sq, m, 32); }
    const float mean = s / (float)N;
    const float var  = sq / (float)N - mean * mean;
    const float inv  = rsqrtf(var + LN_EPS);
    float* crow = Cm + (size_t)(m0 + row) * ldc;
#pragma unroll
    for (int j = 0; j < N / 16; ++j) {
        const int jj = idx + 16 * j;
        const float x = hbuf[row * N + jj];
        crow[jj] = (x - mean) * inv * gamma[jj] + beta[jj];
    }
}

// ---------------------------------------------------------------------------
// Kernel 3a: gather the matching expert row:
//   feature[b, H+n] = spec_out[D[b]][b][n]
// ---------------------------------------------------------------------------
__global__ void spec_gather(const float* __restrict__ spec_out, const int* __restrict__ D,
                            float* __restrict__ feat) {
    const int b   = blockIdx.x;
    const int tid = threadIdx.x;                    // 0..383
    const int d   = D[b];
    feat[(size_t)b * (2 * HDIM) + HDIM + tid] =
        spec_out[((size_t)d * BDIM + b) * HDIM + tid];
}

// ---------------------------------------------------------------------------
// Kernel 3b (fallback if ws too small): per-sample expert matvec + LN
// ---------------------------------------------------------------------------
__global__ void spec_expert(const float* __restrict__ bert, const int* __restrict__ D,
                            const float* __restrict__ spec_W, const float* __restrict__ spec_b,
                            const float* __restrict__ spec_g, const float* __restrict__ spec_be,
                            float* __restrict__ feat) {
    __shared__ float xs[EDIM];
    __shared__ float rs[512];
    __shared__ float rq[512];
    const int tid = threadIdx.x;           // 0..383 == output column
    const int b   = blockIdx.x;

    for (int e = tid; e < EDIM; e += HDIM) xs[e] = bert[(size_t)b * EDIM + e];
    if (tid < 128) { rs[HDIM + tid] = 0.f; rq[HDIM + tid] = 0.f; }
    __syncthreads();

    const int d = D[b];
    const float* Wb = spec_W + (size_t)d * EDIM * HDIM;
    float acc = spec_b[(size_t)d * HDIM + tid];
    for (int e = 0; e < EDIM; ++e) acc += xs[e] * Wb[(size_t)e * HDIM + tid];
    acc = leaky(acc);

    rs[tid] = acc; rq[tid] = acc * acc;
    __syncthreads();
    for (int off = 256; off > 0; off >>= 1) {
        if (tid < off) { rs[tid] += rs[tid + off]; rq[tid] += rq[tid + off]; }
        __syncthreads();
    }
    const float mean = rs[0] / (float)HDIM;
    const float var  = rq[0] / (float)HDIM - mean * mean;
    const float inv  = rsqrtf(var + LN_EPS);
    feat[(size_t)b * (2 * HDIM) + HDIM + tid] =
        (acc - mean) * inv * spec_g[(size_t)d * HDIM + tid] + spec_be[(size_t)d * HDIM + tid];
}

// ---------------------------------------------------------------------------
// Kernel 4: classifier head  out[b,0:2] = feature[b,:] @ cls_W + cls_b
// ---------------------------------------------------------------------------
__global__ void cls_head(const float* __restrict__ feat, const float* __restrict__ cls_W,
                         const float* __restrict__ cls_b, float* __restrict__ out) {
    const int b = blockIdx.x, lane = threadIdx.x;
    float a0 = 0.f, a1 = 0.f;
    for (int e = lane; e < 2 * HDIM; e += 32) {
        const float f = feat[(size_t)b * (2 * HDIM) + e];
        a0 += f * cls_W[2 * e];
        a1 += f * cls_W[2 * e + 1];
    }
    for (int m = 16; m >= 1; m >>= 1) { a0 += __shfl_xor(a0, m, 32); a1 += __shfl_xor(a1, m, 32); }
    if (lane == 0) { out[2 * b] = a0 + cls_b[0]; out[2 * b + 1] = a1 + cls_b[1]; }
}

// ---------------------------------------------------------------------------
// Kernel 5: domain head  dp[b,n] = leaky(d[b,:]) @ dom_W2 + dom_b2
// ---------------------------------------------------------------------------
__global__ void dom_head(const float* __restrict__ dvec, const float* __restrict__ dom_W2,
                         const float* __restrict__ dom_b2, float* __restrict__ dp) {
    const int b = blockIdx.x;
    const int w = threadIdx.x >> 5, lane = threadIdx.x & 31;
    float a = 0.f;
    for (int e = lane; e < HDIM; e += 32) {
        float v = dvec[(size_t)b * HDIM + e];
        v = leaky(v);
        a += v * dom_W2[(size_t)e * NDOM + w];
    }
    for (int m = 16; m >= 1; m >>= 1) a += __shfl_xor(a, m, 32);
    if (lane == 0) dp[(size_t)b * NDOM + w] = a + dom_b2[w];
}

// ---------------------------------------------------------------------------
extern "C" void kernel_launch(void* const* d_in, const int* in_sizes, int n_in,
                              void* d_out, int out_size, void* d_ws, size_t ws_size,
                              hipStream_t stream) {
    (void)in_sizes; (void)n_in; (void)out_size;
    const float* input    = (const float*)d_in[0];
    const int*   D        = (const int*)  d_in[1];
    const float* w_att    = (const float*)d_in[2];
    const float* shared_W = (const float*)d_in[3];
    const float* shared_b = (const float*)d_in[4];
    const float* shared_g = (const float*)d_in[5];
    const float* shared_be= (const float*)d_in[6];
    const float* spec_W   = (const float*)d_in[7];
    const float* spec_b   = (const float*)d_in[8];
    const float* spec_g   = (const float*)d_in[9];
    const float* spec_be  = (const float*)d_in[10];
    const float* dec_W1   = (const float*)d_in[11];
    const float* dec_b1   = (const float*)d_in[12];
    const float* dec_g1   = (const float*)d_in[13];
    const float* dec_be1  = (const float*)d_in[14];
    const float* dec_W2   = (const float*)d_in[15];
    const float* dec_b2   = (const float*)d_in[16];
    const float* dec_g2   = (const float*)d_in[17];
    const float* dec_be2  = (const float*)d_in[18];
    const float* cls_W    = (const float*)d_in[19];
    const float* cls_b    = (const float*)d_in[20];
    const float* dom_W1   = (const float*)d_in[21];
    const float* dom_b1   = (const float*)d_in[22];
    const float* dom_g1   = (const float*)d_in[23];
    const float* dom_be1  = (const float*)d_in[24];
    const float* dom_W2   = (const float*)d_in[25];
    const float* dom_b2   = (const float*)d_in[26];

    float* out      = (float*)d_out;
    float* out_cls  = out;                                      // [256,2]
    float* out_rec  = out + 512;                                // [256,768]
    float* out_bert = out + 512 + BDIM * EDIM;                  // [256,768]
    float* out_dom  = out + 512 + 2 * BDIM * EDIM;              // [256,9]

    float* ws      = (float*)d_ws;
    float* feature = ws;                                        // [256, 768] = [shared|specific]
    float* rec     = feature + (size_t)BDIM * 2 * HDIM;         // [256, 384]
    float* domd    = rec + (size_t)BDIM * HDIM;                 // [256, 384]
    float* specout = domd + (size_t)BDIM * HDIM;                // [9, 256, 384] (optional)

    const size_t ws_need_full =
        ((size_t)BDIM * 2 * HDIM + 2 * (size_t)BDIM * HDIM +
         (size_t)NDOM * BDIM * HDIM) * sizeof(float);

    // 1. attention pooling -> bert_feature (third output, also GEMM input)
    attn_pool<<<BDIM, 256, 0, stream>>>(input, w_att, out_bert);

    // 2. shared branch -> feature[:, 0:H]
    gemm_bias_leaky_ln<EDIM, HDIM><<<BDIM / 16, 256, 0, stream>>>(
        out_bert, EDIM, shared_W, shared_b, shared_g, shared_be, feature, 2 * HDIM);

    // 3. expert branch -> feature[:, H:2H]
    if (ws_size >= ws_need_full) {
        // 9 experts: run each expert GEMM over all rows on the matrix pipe
        // (weights read once ~10.6 MB total instead of 302 MB), then gather.
        for (int d = 0; d < NDOM; ++d) {
            gemm_bias_leaky_ln<EDIM, HDIM><<<BDIM / 16, 256, 0, stream>>>(
                out_bert, EDIM,
                spec_W + (size_t)d * EDIM * HDIM,
                spec_b + (size_t)d * HDIM, spec_g + (size_t)d * HDIM,
                spec_be + (size_t)d * HDIM,
                specout + (size_t)d * BDIM * HDIM, HDIM);
        }
        spec_gather<<<BDIM, HDIM, 0, stream>>>(specout, D, feature);
    } else {
        spec_expert<<<BDIM, HDIM, 0, stream>>>(out_bert, D, spec_W, spec_b, spec_g, spec_be, feature);
    }

    // 4. decoder stage 1: rec = mlp(feature, dec_W1)
    gemm_bias_leaky_ln<2 * HDIM, HDIM><<<BDIM / 16, 256, 0, stream>>>(
        feature, 2 * HDIM, dec_W1, dec_b1, dec_g1, dec_be1, rec, HDIM);

    // 5. decoder stage 2: rec_feature = mlp(rec, dec_W2)  (second output)
    gemm_bias_leaky_ln<HDIM, EDIM><<<BDIM / 16, 256, 0, stream>>>(
        rec, HDIM, dec_W2, dec_b2, dec_g2, dec_be2, out_rec, EDIM);

    // 6. domain stage 1: d = mlp(shared_feature, dom_W1)  (reads feature cols 0..H-1)
    gemm_bias_leaky_ln<HDIM, HDIM><<<BDIM / 16, 256, 0, stream>>>(
        feature, 2 * HDIM, dom_W1, dom_b1, dom_g1, dom_be1, domd, HDIM);

    // 7. classifier head (first output)
    cls_head<<<BDIM, 32, 0, stream>>>(feature, cls_W, cls_b, out_cls);

    // 8. domain head (fourth output)
    dom_head<<<BDIM, 32 * NDOM, 0, stream>>>(domd, dom_W2, dom_b2, out_dom);
}